// GCNLayer_7868380086997
// MI455X (gfx1250) — compile-verified
//
#include <hip/hip_runtime.h>

typedef __attribute__((ext_vector_type(2))) float v2f;
typedef __attribute__((ext_vector_type(8))) float v8f;

#define F_IN 128
#define XDIM 256          // 2*F_IN, also 2*F_OUT (GEMM1 output width)
#define F_OUT 128
#define LDS_STRIDE 260    // 256 + 4 dword pad: lane m -> bank (4m+c)%64, conflict-free

// ---------------------------------------------------------------------------
// Kernel 1: norm_h = h * norm (row-broadcast), and zero the agg accumulator.
// One float4 per thread.
// ---------------------------------------------------------------------------
__global__ void gcn_scale_zero_kernel(const float* __restrict__ h,
                                      const float* __restrict__ norm,
                                      float* __restrict__ norm_h,
                                      float* __restrict__ agg,
                                      int n_nodes) {
  int t = blockIdx.x * blockDim.x + threadIdx.x;   // float4 index
  int total = n_nodes * (F_IN / 4);
  if (t >= total) return;
  int node = t / (F_IN / 4);
  float s = norm[node];
  float4 v = ((const float4*)h)[t];
  v.x *= s; v.y *= s; v.z *= s; v.w *= s;
  ((float4*)norm_h)[t] = v;
  ((float4*)agg)[t] = make_float4(0.f, 0.f, 0.f, 0.f);
}

// ---------------------------------------------------------------------------
// Kernel 2: edge scatter. One wave (32 lanes) per edge; each lane moves a
// float4 of features and issues 4 native f32 atomics. norm_h and agg are
// L2-resident (51 MB each << 192 MB L2), so atomics resolve at L2.
// ---------------------------------------------------------------------------
__global__ void gcn_scatter_kernel(const float* __restrict__ norm_h,
                                   const int* __restrict__ src,
                                   const int* __restrict__ dst,
                                   float* __restrict__ agg,
                                   int n_edges) {
  long long t = (long long)blockIdx.x * blockDim.x + threadIdx.x;
  int edge = (int)(t >> 5);
  if (edge >= n_edges) return;
  int lane = (int)(t & 31);
  int s = src[edge];
  int d = dst[edge];
  float4 v = ((const float4*)(norm_h + (long long)s * F_IN))[lane];
  float* base = agg + (long long)d * F_IN + lane * 4;
  unsafeAtomicAdd(base + 0, v.x);   // global_atomic_add_f32, no return
  unsafeAtomicAdd(base + 1, v.y);
  unsafeAtomicAdd(base + 2, v.z);
  unsafeAtomicAdd(base + 3, v.w);
}

// ---------------------------------------------------------------------------
// Kernel 3 (fused): per 16-node tile:
//   xin = concat(h, agg*norm)            [16 x 256]  (LDS)
//   x1  = relu(xin @ W1)                 [16 x 256]  (WMMA f32 16x16x4, LDS)
//   x1n = x1 * rsqrt(max(rowsumsq, eps))
//   out = relu(x1n @ W2)                 [16 x 128]  (WMMA f32 16x16x4)
// 4 waves / block; each wave owns 16x16 C tiles (4 tiles GEMM1, 2 tiles GEMM2).
// ---------------------------------------------------------------------------
__global__ __launch_bounds__(128)
void gcn_mlp_kernel(const float* __restrict__ h,
                    const float* __restrict__ norm,
                    const float* __restrict__ agg,
                    const float* __restrict__ W1,   // [256 x 256] row-major
                    const float* __restrict__ W2,   // [256 x 128] row-major
                    float* __restrict__ out,        // [N x 128]
                    int n_nodes) {
  __shared__ float xin[16 * LDS_STRIDE];
  __shared__ float x1[16 * LDS_STRIDE];
  __shared__ float rstd[16];

  const int tid  = threadIdx.x;
  const int wave = tid >> 5;
  const int lane = tid & 31;
  const int g    = lane >> 4;     // half-wave group: selects K pair / M offset
  const int ln   = lane & 15;     // M (A) or N (B,C) index within tile
  const int node0 = blockIdx.x * 16;

  // ---- Stage input tile into LDS: [r][0:128] = h, [r][128:256] = agg*norm.
  for (int i = tid; i < 16 * (XDIM / 4); i += 128) {
    int r  = i / (XDIM / 4);
    int c4 = i % (XDIM / 4);
    int node = node0 + r;
    float4 v = make_float4(0.f, 0.f, 0.f, 0.f);
    if (node < n_nodes) {
      if (c4 < F_IN / 4) {
        v = ((const float4*)(h + (long long)node * F_IN))[c4];
      } else {
        float s = norm[node];
        v = ((const float4*)(agg + (long long)node * F_IN))[c4 - F_IN / 4];
        v.x *= s; v.y *= s; v.z *= s; v.w *= s;
      }
    }
    float* p = &xin[r * LDS_STRIDE + c4 * 4];
    p[0] = v.x; p[1] = v.y; p[2] = v.z; p[3] = v.w;
  }
  __syncthreads();

  // ---- GEMM1: x1 = relu(xin @ W1). Each wave: 4 column tiles of 16.
  for (int t4 = 0; t4 < 4; ++t4) {
    const int ncol0 = (wave * 4 + t4) * 16;
    v8f acc = {};
    #pragma unroll 4
    for (int k0 = 0; k0 < XDIM; k0 += 4) {
      const int ka = k0 + 2 * g;          // K pair for this half-wave
      v2f a, b;
      a.x = xin[ln * LDS_STRIDE + ka];
      a.y = xin[ln * LDS_STRIDE + ka + 1];
      b.x = W1[ka * XDIM + ncol0 + ln];
      b.y = W1[(ka + 1) * XDIM + ncol0 + ln];
      acc = __builtin_amdgcn_wmma_f32_16x16x4_f32(
          /*neg_a=*/false, a, /*neg_b=*/false, b,
          /*c_mod=*/(short)0, acc, /*reuse_a=*/false, /*reuse_b=*/false);
    }
    // ReLU + store C tile to LDS (C layout: vgpr r, lane l -> m=r+8g, n=ln)
    #pragma unroll
    for (int r = 0; r < 8; ++r) {
      int m = r + 8 * g;
      x1[m * LDS_STRIDE + ncol0 + ln] = fmaxf(acc[r], 0.f);
    }
  }
  __syncthreads();

  // ---- Row L2 norm factors.
  if (tid < 16) {
    float sq = 0.f;
    for (int c = 0; c < XDIM; ++c) {
      float v = x1[tid * LDS_STRIDE + c];
      sq += v * v;
    }
    rstd[tid] = rsqrtf(fmaxf(sq, 1e-12f));
  }
  __syncthreads();

  // ---- GEMM2: out = relu((x1*rstd) @ W2). Each wave: 2 column tiles of 16.
  const float rs = rstd[ln];              // row scale for this lane's M rows
  for (int t2 = 0; t2 < 2; ++t2) {
    const int ncol0 = (wave * 2 + t2) * 16;
    v8f acc = {};
    #pragma unroll 4
    for (int k0 = 0; k0 < XDIM; k0 += 4) {
      const int ka = k0 + 2 * g;
      v2f a, b;
      a.x = x1[ln * LDS_STRIDE + ka] * rs;
      a.y = x1[ln * LDS_STRIDE + ka + 1] * rs;
      b.x = W2[ka * F_OUT + ncol0 + ln];
      b.y = W2[(ka + 1) * F_OUT + ncol0 + ln];
      acc = __builtin_amdgcn_wmma_f32_16x16x4_f32(
          false, a, false, b, (short)0, acc, false, false);
    }
    #pragma unroll
    for (int r = 0; r < 8; ++r) {
      int m = r + 8 * g;
      int node = node0 + m;
      if (node < n_nodes)
        out[(long long)node * F_OUT + ncol0 + ln] = fmaxf(acc[r], 0.f);
    }
  }
}

// ---------------------------------------------------------------------------
// Launch. Inputs: h [N*128], norm [N], weight [256*256], weight2 [256*128],
// src [E], dst [E]. Output: [N*128] f32. Workspace: norm_h + agg (102.4 MB).
// ---------------------------------------------------------------------------
extern "C" void kernel_launch(void* const* d_in, const int* in_sizes, int n_in,
                              void* d_out, int out_size, void* d_ws, size_t ws_size,
                              hipStream_t stream) {
  const float* h    = (const float*)d_in[0];
  const float* norm = (const float*)d_in[1];
  const float* W1   = (const float*)d_in[2];
  const float* W2   = (const float*)d_in[3];
  const int*   src  = (const int*)d_in[4];
  const int*   dst  = (const int*)d_in[5];
  float* out = (float*)d_out;

  const int n_nodes = in_sizes[1];   // norm is [N,1]
  const int n_edges = in_sizes[4];

  float* norm_h = (float*)d_ws;
  float* agg    = norm_h + (size_t)n_nodes * F_IN;

  {
    int total = n_nodes * (F_IN / 4);
    int blocks = (total + 255) / 256;
    gcn_scale_zero_kernel<<<blocks, 256, 0, stream>>>(h, norm, norm_h, agg, n_nodes);
  }
  {
    long long total = (long long)n_edges * 32;
    int blocks = (int)((total + 255) / 256);
    gcn_scatter_kernel<<<blocks, 256, 0, stream>>>(norm_h, src, dst, agg, n_edges);
  }
  {
    int blocks = (n_nodes + 15) / 16;
    gcn_mlp_kernel<<<blocks, 128, 0, stream>>>(h, norm, agg, W1, W2, out, n_nodes);
  }
}